// ComplEx_29231547417249
// MI455X (gfx1250) — compile-verified
//
#include <hip/hip_runtime.h>

typedef float v2f __attribute__((ext_vector_type(2)));
typedef float v4f __attribute__((ext_vector_type(4)));
typedef float v8f __attribute__((ext_vector_type(8)));

#define EMB_D     128
#define ROW       256          // 2*EMB_D floats per node row
#define N_EDGES   1000000
#define TILES     (N_EDGES / 16)
#define WAVES_PER_BLOCK 8

// h_e = sum_d  hr*rr*tr + hi*rr*ti + hr*ri*ti - hi*ri*tr
//     = sum_d  hr*u + hi*v   with  u = rr*tr + ri*ti,  v = rr*ti - ri*tr
//
// One wave = 16 edges. A-matrix rows = head chunks (16x4 f32), B-matrix cols =
// rotated tail chunks (4x16 f32), accumulate via v_wmma_f32_16x16x4_f32.
// Diagonal of the 16x16 f32 accumulator = the 16 edge scores (full f32 math).
__global__ __launch_bounds__(256) void complex_score_wmma(
    const float* __restrict__ node_emb,
    const float* __restrict__ rel_emb,
    const int*   __restrict__ edge_idx,   // [2, N_EDGES] row-major
    float*       __restrict__ out)
{
    __shared__ float s_rel[ROW];
    s_rel[threadIdx.x] = rel_emb[threadIdx.x];   // blockDim == 256 == ROW
    __syncthreads();

    const int lane = threadIdx.x & 31;
    const int wave = threadIdx.x >> 5;
    const int tile = blockIdx.x * WAVES_PER_BLOCK + wave;
    if (tile >= TILES) return;                   // wave-uniform: EXEC stays all-1s

    const int row  = lane & 15;                  // edge within tile (A row M / B col N)
    const int half = lane >> 4;                  // K half-wave subgroup
    const int e    = tile * 16 + row;

    const long long hbase = (long long)edge_idx[e]           * ROW;
    const long long tbase = (long long)edge_idx[N_EDGES + e] * ROW;
    const float* __restrict__ hrow = node_emb + hbase;
    const float* __restrict__ trow = node_emb + tbase;

    v8f acc1 = {0.f, 0.f, 0.f, 0.f, 0.f, 0.f, 0.f, 0.f};
    v8f acc2 = {0.f, 0.f, 0.f, 0.f, 0.f, 0.f, 0.f, 0.f};

    // K=8 elements per iteration; float4 (global_load_b128) loads per matrix.
    // Element->K labeling is identical for A and B, so the sum is exact even
    // though each WMMA covers a permuted K subset {k,k+1,k+4,k+5} / {k+2,...}.
    #pragma unroll 4
    for (int k = 0; k < EMB_D; k += 8) {
        const int ko = k + 4 * half;
        const v4f hr = *(const v4f*)(hrow + ko);
        const v4f hi = *(const v4f*)(hrow + EMB_D + ko);
        const v4f tr = *(const v4f*)(trow + ko);
        const v4f ti = *(const v4f*)(trow + EMB_D + ko);
        const v4f rr = *(const v4f*)(s_rel + ko);
        const v4f ri = *(const v4f*)(s_rel + EMB_D + ko);

        const v4f u = rr * tr + ri * ti;   // pairs with head_re
        const v4f v = rr * ti - ri * tr;   // pairs with head_im

        const v2f a0 = {hr.x, hr.y}, a1 = {hr.z, hr.w};
        const v2f b0 = {u.x,  u.y},  b1 = {u.z,  u.w};
        const v2f c0 = {hi.x, hi.y}, c1 = {hi.z, hi.w};
        const v2f w0 = {v.x,  v.y},  w1 = {v.z,  v.w};

        acc1 = __builtin_amdgcn_wmma_f32_16x16x4_f32(false, a0, false, b0,
                                                     (short)0, acc1, false, false);
        acc2 = __builtin_amdgcn_wmma_f32_16x16x4_f32(false, c0, false, w0,
                                                     (short)0, acc2, false, false);
        acc1 = __builtin_amdgcn_wmma_f32_16x16x4_f32(false, a1, false, b1,
                                                     (short)0, acc1, false, false);
        acc2 = __builtin_amdgcn_wmma_f32_16x16x4_f32(false, c1, false, w1,
                                                     (short)0, acc2, false, false);
    }

    const v8f d = acc1 + acc2;

    // Diagonal extraction per C/D layout:
    //   e < 8 : lane e,      VGPR e        e >= 8 : lane e+16,  VGPR e-8
    const int sel = (lane < 8) ? lane : (lane - 24);
    float val = d[0];
    val = (sel == 1) ? d[1] : val;
    val = (sel == 2) ? d[2] : val;
    val = (sel == 3) ? d[3] : val;
    val = (sel == 4) ? d[4] : val;
    val = (sel == 5) ? d[5] : val;
    val = (sel == 6) ? d[6] : val;
    val = (sel == 7) ? d[7] : val;

    const bool active = (lane < 8) || (lane >= 24);
    if (active) {
        const int oe = tile * 16 + ((lane < 8) ? lane : (lane - 16));
        out[oe] = val;
    }
}

extern "C" void kernel_launch(void* const* d_in, const int* in_sizes, int n_in,
                              void* d_out, int out_size, void* d_ws, size_t ws_size,
                              hipStream_t stream) {
    const float* node_emb = (const float*)d_in[0];   // [200000, 256] f32
    const float* rel_emb  = (const float*)d_in[1];   // [1, 256] f32
    const int*   eidx     = (const int*)d_in[2];     // [2, 1000000] int
    float*       out      = (float*)d_out;           // [1000000] f32

    const int grid = (TILES + WAVES_PER_BLOCK - 1) / WAVES_PER_BLOCK;  // 7813
    hipLaunchKernelGGL(complex_score_wmma, dim3(grid), dim3(256), 0, stream,
                       node_emb, rel_emb, eidx, out);
}